// GCN_52544629899901
// MI455X (gfx1250) — compile-verified
//
#include <hip/hip_runtime.h>

// GCN 2-layer: y = D^-1/2 (A+I) D^-1/2 (x W^T) + b, twice (ReLU between).
// Dense GEMMs via V_WMMA_F32_16X16X4_F32 (full fp32, matrix pipe).
// Edge scatter is the bottleneck: L2-resident (all intermediates < 192MB L2),
// float4 gathers + hardware global_atomic_add_f32.

typedef float v2f __attribute__((ext_vector_type(2)));
typedef float v8f __attribute__((ext_vector_type(8)));

constexpr int IN_C  = 256;
constexpr int HID   = 128;
constexpr int OUT_C = 64;

// ---------------- WMMA GEMM: C[M,NC] = A[M,K] * W[NC,K]^T (all row-major) ---
// One wave per 16x16 C tile. f32 WMMA, K step = 4.
// A frag (16x4 f32): lanes 0-15 hold row M=l, VGPR0/1 = K+0/K+1;
//                    lanes 16-31 hold row M=l-16, VGPR0/1 = K+2/K+3.
// B frag (4x16 f32): lane half selects K pair, lane%16 selects N column.
//                    B[k][n] == W[n*K + k] since B = W^T.
// C/D (16x16 f32):   VGPR v, lane l -> row = v + (l/16)*8, col = l%16.
template <int K, int NC>
__global__ __launch_bounds__(256) void gemm_xwt_wmma(
    const float* __restrict__ A, const float* __restrict__ W,
    float* __restrict__ C, int M) {
  const int wave = (blockIdx.x * blockDim.x + threadIdx.x) >> 5;
  constexpr int tilesN = NC / 16;
  const int tm = wave / tilesN;
  const int tn = wave % tilesN;
  if (tm * 16 >= M) return;  // wave-uniform: EXEC stays all-ones for WMMA
  const int lane = threadIdx.x & 31;
  const int lh = lane >> 4;   // 0 or 1 (K-half)
  const int lm = lane & 15;   // row (A) / col (B) within tile

  const float* a0 = A + (size_t)(tm * 16 + lm) * K + lh * 2;
  const float* b0 = W + (size_t)(tn * 16 + lm) * K + lh * 2;

  v8f acc = {0.f, 0.f, 0.f, 0.f, 0.f, 0.f, 0.f, 0.f};
#pragma unroll 8
  for (int k = 0; k < K; k += 4) {
    v2f a; a.x = a0[k]; a.y = a0[k + 1];
    v2f b; b.x = b0[k]; b.y = b0[k + 1];
    acc = __builtin_amdgcn_wmma_f32_16x16x4_f32(
        /*neg_a=*/false, a, /*neg_b=*/false, b,
        /*c_mod=*/(short)0, acc, /*reuse_a=*/false, /*reuse_b=*/false);
  }

  float* c0 = C + (size_t)(tm * 16 + lh * 8) * NC + tn * 16 + lm;
#pragma unroll
  for (int v = 0; v < 8; ++v) c0[(size_t)v * NC] = acc[v];
}

// ---------------- degree: deg[col] += ew ----------------------------------
__global__ __launch_bounds__(256) void degree_kernel(
    const long long* __restrict__ ei, const float* __restrict__ ew,
    float* __restrict__ deg, int E) {
  int e = blockIdx.x * blockDim.x + threadIdx.x;
  if (e >= E) return;
  unsafeAtomicAdd(deg + ei[(long long)E + e], ew[e]);
}

// deg -> deg^-1/2 in place (deg >= 1 always after +1 self loop)
__global__ __launch_bounds__(256) void dinv_kernel(float* __restrict__ deg,
                                                   int N) {
  int i = blockIdx.x * blockDim.x + threadIdx.x;
  if (i >= N) return;
  deg[i] = 1.0f / sqrtf(deg[i] + 1.0f);
}

// ---------------- edge scatter, 128 feats: one wave/edge, float4/lane -----
__global__ __launch_bounds__(256) void scatter_edges_128(
    const long long* __restrict__ ei, const float* __restrict__ ew,
    const float* __restrict__ dinv, const float* __restrict__ src,
    float* __restrict__ dst, int E) {
  long long t = (long long)blockIdx.x * blockDim.x + threadIdx.x;
  int e = (int)(t >> 5);
  if (e >= E) return;
  int lane = threadIdx.x & 31;
  long long r = ei[e];
  long long c = ei[(long long)E + e];
  float norm = dinv[r] * ew[e] * dinv[c];
  float4 m = ((const float4*)(src + r * 128))[lane];  // coalesced b128 gather
  float* d = dst + c * 128 + lane * 4;
  unsafeAtomicAdd(d + 0, norm * m.x);
  unsafeAtomicAdd(d + 1, norm * m.y);
  unsafeAtomicAdd(d + 2, norm * m.z);
  unsafeAtomicAdd(d + 3, norm * m.w);
}

// ---------------- edge scatter, 64 feats: one wave/edge, float2/lane ------
__global__ __launch_bounds__(256) void scatter_edges_64(
    const long long* __restrict__ ei, const float* __restrict__ ew,
    const float* __restrict__ dinv, const float* __restrict__ src,
    float* __restrict__ dst, int E) {
  long long t = (long long)blockIdx.x * blockDim.x + threadIdx.x;
  int e = (int)(t >> 5);
  if (e >= E) return;
  int lane = threadIdx.x & 31;
  long long r = ei[e];
  long long c = ei[(long long)E + e];
  float norm = dinv[r] * ew[e] * dinv[c];
  float2 m = ((const float2*)(src + r * 64))[lane];
  float* d = dst + c * 64 + lane * 2;
  unsafeAtomicAdd(d + 0, norm * m.x);
  unsafeAtomicAdd(d + 1, norm * m.y);
}

// h = relu(agg + dinv^2 * xw + b)   (128 feats, in place on agg)
__global__ __launch_bounds__(256) void fuse_selfloop_relu_128(
    float* __restrict__ agg, const float* __restrict__ xw,
    const float* __restrict__ dinv, const float* __restrict__ b, int N) {
  long long t = (long long)blockIdx.x * blockDim.x + threadIdx.x;
  int i = (int)(t >> 5);
  if (i >= N) return;
  int lane = threadIdx.x & 31;
  float di = dinv[i];
  float s = di * di;
  float4 a = ((const float4*)(agg + (size_t)i * 128))[lane];
  float4 x = ((const float4*)(xw + (size_t)i * 128))[lane];
  float4 bb = ((const float4*)b)[lane];
  float4 r;
  r.x = fmaxf(fmaf(s, x.x, a.x) + bb.x, 0.f);
  r.y = fmaxf(fmaf(s, x.y, a.y) + bb.y, 0.f);
  r.z = fmaxf(fmaf(s, x.z, a.z) + bb.z, 0.f);
  r.w = fmaxf(fmaf(s, x.w, a.w) + bb.w, 0.f);
  ((float4*)(agg + (size_t)i * 128))[lane] = r;
}

// out = out + dinv^2 * hw + b   (64 feats; out already holds edge agg)
__global__ __launch_bounds__(256) void fuse_out_64(
    float* __restrict__ out, const float* __restrict__ hw,
    const float* __restrict__ dinv, const float* __restrict__ b, int N) {
  long long t = (long long)blockIdx.x * blockDim.x + threadIdx.x;
  int i = (int)(t >> 4);
  if (i >= N) return;
  int lane = threadIdx.x & 15;
  float di = dinv[i];
  float s = di * di;
  float4 o = ((const float4*)(out + (size_t)i * 64))[lane];
  float4 x = ((const float4*)(hw + (size_t)i * 64))[lane];
  float4 bb = ((const float4*)b)[lane];
  o.x = fmaf(s, x.x, o.x) + bb.x;
  o.y = fmaf(s, x.y, o.y) + bb.y;
  o.z = fmaf(s, x.z, o.z) + bb.z;
  o.w = fmaf(s, x.w, o.w) + bb.w;
  ((float4*)(out + (size_t)i * 64))[lane] = o;
}

extern "C" void kernel_launch(void* const* d_in, const int* in_sizes, int n_in,
                              void* d_out, int out_size, void* d_ws,
                              size_t ws_size, hipStream_t stream) {
  const float*     x  = (const float*)d_in[0];
  const long long* ei = (const long long*)d_in[1];  // [2,E] int64
  const float*     ew = (const float*)d_in[2];
  const float*     W1 = (const float*)d_in[4];      // [HID, IN_C]
  const float*     b1 = (const float*)d_in[5];
  const float*     W2 = (const float*)d_in[6];      // [OUT_C, HID]
  const float*     b2 = (const float*)d_in[7];
  float* out = (float*)d_out;

  const int N = in_sizes[3];  // node_type has N elements
  const int E = in_sizes[2];  // edge_weight has E elements

  // workspace layout
  float* dinv = (float*)d_ws;                 // N (deg -> dinv in place)
  float* xw1  = dinv + N;                     // N*HID
  float* agg1 = xw1 + (size_t)N * HID;        // N*HID (becomes h)
  float* hw2  = agg1 + (size_t)N * HID;       // N*OUT_C

  // per-call zero init (atomic accumulation targets)
  hipMemsetAsync(dinv, 0, sizeof(float) * (size_t)N, stream);
  hipMemsetAsync(agg1, 0, sizeof(float) * (size_t)N * HID, stream);
  hipMemsetAsync(out,  0, sizeof(float) * (size_t)N * OUT_C, stream);

  // degrees, then dinv
  degree_kernel<<<(E + 255) / 256, 256, 0, stream>>>(ei, ew, dinv, E);
  dinv_kernel<<<(N + 255) / 256, 256, 0, stream>>>(dinv, N);

  // layer 1: xw1 = x @ W1^T  (WMMA)
  {
    int waves = ((N + 15) / 16) * (HID / 16);
    gemm_xwt_wmma<IN_C, HID>
        <<<((long long)waves * 32 + 255) / 256, 256, 0, stream>>>(x, W1, xw1, N);
  }
  // edge aggregation into agg1
  {
    long long th = (long long)E * 32;
    scatter_edges_128<<<(int)((th + 255) / 256), 256, 0, stream>>>(
        ei, ew, dinv, xw1, agg1, E);
  }
  // self-loop + bias + relu -> h (in agg1)
  {
    long long th = (long long)N * 32;
    fuse_selfloop_relu_128<<<(int)((th + 255) / 256), 256, 0, stream>>>(
        agg1, xw1, dinv, b1, N);
  }
  // layer 2: hw2 = h @ W2^T  (WMMA)
  {
    int waves = ((N + 15) / 16) * (OUT_C / 16);
    gemm_xwt_wmma<HID, OUT_C>
        <<<((long long)waves * 32 + 255) / 256, 256, 0, stream>>>(agg1, W2, hw2, N);
  }
  // edge aggregation directly into out
  {
    long long th = (long long)E * 32;
    scatter_edges_64<<<(int)((th + 255) / 256), 256, 0, stream>>>(
        ei, ew, dinv, hw2, out, E);
  }
  // self-loop + bias
  {
    long long th = (long long)N * 16;
    fuse_out_64<<<(int)((th + 255) / 256), 256, 0, stream>>>(out, hw2, dinv, b2, N);
  }
}